// VQGAN_CLIP_10634339024943
// MI455X (gfx1250) — compile-verified
//
#include <hip/hip_runtime.h>

typedef __attribute__((ext_vector_type(2))) float v2f;
typedef __attribute__((ext_vector_type(8))) float v8f;

#define DIM      256
#define NCODES   16384
#define NPIX     4096          // 4*32*32
#define MB       64            // pixels per block
#define NSLICE   4             // N split across blocks
#define KSTEPS   (DIM / 4)     // 64 WMMA K-steps of 4
#define APITCH   257           // LDS pitch padding (bank-conflict avoidance)

// ---------------------------------------------------------------------------
// Kernel 1: per-code squared norms ||c||^2
// ---------------------------------------------------------------------------
__global__ void __launch_bounds__(256) cnorm_kernel(const float* __restrict__ cb,
                                                    float* __restrict__ cnorm) {
    int n = blockIdx.x * blockDim.x + threadIdx.x;
    const float4* row = (const float4*)(cb + (size_t)n * DIM);
    float s = 0.0f;
#pragma unroll 8
    for (int i = 0; i < DIM / 4; ++i) {
        float4 v = row[i];
        s += v.x * v.x + v.y * v.y + v.z * v.z + v.w * v.w;
    }
    cnorm[n] = s;
}

// ---------------------------------------------------------------------------
// Kernel 2: fused distance GEMM (fp32 WMMA) + partial argmin over an N-slice.
// Block = 64 pixels x 4096 codes. 8 waves, each wave owns 512 codes,
// processed as 16 iterations of two 16-code N-tiles (8 accumulators) so each
// A fragment (LDS) feeds 2 WMMAs.
// d(p,n) = ||c_n||^2 - 2 * <x_p, c_n>   (||x_p||^2 constant per pixel -> drop)
// ---------------------------------------------------------------------------
__global__ void __launch_bounds__(256) vq_argmin_kernel(
    const float* __restrict__ latent,   // [4,256,32,32] NCHW
    const float* __restrict__ cb,       // [16384,256]
    const float* __restrict__ cnorm,    // [16384]
    float* __restrict__ pval,           // [NSLICE, NPIX]
    int*   __restrict__ pidx)           // [NSLICE, NPIX]
{
    __shared__ float A[MB * APITCH];    // A[m][k], pitch-padded
    __shared__ float wval[8 * MB];
    __shared__ int   widx[8 * MB];

    const int t     = threadIdx.x;
    const int bx    = blockIdx.x;
    const int mblk  = bx >> 2;          // 0..63
    const int nsl   = bx & (NSLICE - 1);
    const int mbase = mblk * MB;        // global pixel base (same batch: 64 | 1024)
    const int b     = mbase >> 10;
    const int hw0   = mbase & 1023;
    const float* lat_b = latent + (size_t)b * DIM * 1024;

    // Cooperative A load: A[i][d] = latent[b][d][hw0+i], b128 over pixel axis.
    {
        int i4 = (t & 15) << 2;          // pixel group 0,4,...,60
        int d0 = t >> 4;                 // 0..15
#pragma unroll 4
        for (int it = 0; it < 16; ++it) {
            int d = d0 + (it << 4);
            float4 v = *(const float4*)&lat_b[(size_t)d * 1024 + hw0 + i4];
            A[(i4 + 0) * APITCH + d] = v.x;
            A[(i4 + 1) * APITCH + d] = v.y;
            A[(i4 + 2) * APITCH + d] = v.z;
            A[(i4 + 3) * APITCH + d] = v.w;
        }
    }
    __syncthreads();

    const int wave  = t >> 5;
    const int lane  = t & 31;
    const int laneN = lane & 15;
    const int half  = lane >> 4;        // A/B K split: lanes 16-31 hold k0+2,k0+3
    const int koff  = half << 1;

    float best[4][8];
    int   bidx[4][8];
#pragma unroll
    for (int mt = 0; mt < 4; ++mt)
#pragma unroll
        for (int i = 0; i < 8; ++i) { best[mt][i] = 3.4e38f; bidx[mt][i] = 0; }

    const int nbase = nsl * (NCODES / NSLICE) + wave * 512;

    for (int nt = 0; nt < 16; ++nt) {
        const int nA = nbase + nt * 32 + laneN;       // first 16-code tile
        const int nB = nA + 16;                       // second 16-code tile
        const float4* browA = (const float4*)(cb + (size_t)nA * DIM);
        const float4* browB = (const float4*)(cb + (size_t)nB * DIM);
        // prefetch next iteration's rows (streams 1KB rows out of L2)
        if (nt + 1 < 16) {
            __builtin_prefetch((const char*)(browA) + 32 * DIM * sizeof(float), 0, 0);
            __builtin_prefetch((const char*)(browB) + 32 * DIM * sizeof(float), 0, 0);
        }
        v8f acc0 = {}, acc1 = {}, acc2 = {}, acc3 = {};
        v8f acc4 = {}, acc5 = {}, acc6 = {}, acc7 = {};
#pragma unroll 4
        for (int k = 0; k < KSTEPS; ++k) {
            float4 a4 = browA[k];                 // code row nA, K = 4k..4k+3
            float4 b4 = browB[k];                 // code row nB
            v2f bbA, bbB;
            bbA[0] = half ? a4.z : a4.x;
            bbA[1] = half ? a4.w : a4.y;
            bbB[0] = half ? b4.z : b4.x;
            bbB[1] = half ? b4.w : b4.y;
            const int kbase = (k << 2) + koff;
            v2f a0 = *(const v2f*)&A[(0 * 16 + laneN) * APITCH + kbase];
            v2f a1 = *(const v2f*)&A[(1 * 16 + laneN) * APITCH + kbase];
            v2f a2 = *(const v2f*)&A[(2 * 16 + laneN) * APITCH + kbase];
            v2f a3 = *(const v2f*)&A[(3 * 16 + laneN) * APITCH + kbase];
            acc0 = __builtin_amdgcn_wmma_f32_16x16x4_f32(false, a0, false, bbA, (short)0, acc0, false, false);
            acc1 = __builtin_amdgcn_wmma_f32_16x16x4_f32(false, a1, false, bbA, (short)0, acc1, false, false);
            acc2 = __builtin_amdgcn_wmma_f32_16x16x4_f32(false, a2, false, bbA, (short)0, acc2, false, false);
            acc3 = __builtin_amdgcn_wmma_f32_16x16x4_f32(false, a3, false, bbA, (short)0, acc3, false, false);
            acc4 = __builtin_amdgcn_wmma_f32_16x16x4_f32(false, a0, false, bbB, (short)0, acc4, false, false);
            acc5 = __builtin_amdgcn_wmma_f32_16x16x4_f32(false, a1, false, bbB, (short)0, acc5, false, false);
            acc6 = __builtin_amdgcn_wmma_f32_16x16x4_f32(false, a2, false, bbB, (short)0, acc6, false, false);
            acc7 = __builtin_amdgcn_wmma_f32_16x16x4_f32(false, a3, false, bbB, (short)0, acc7, false, false);
        }
        const float cnA = cnorm[nA];
        const float cnB = cnorm[nB];
#pragma unroll
        for (int i = 0; i < 8; ++i) {
            float d0 = cnA - 2.0f * acc0[i];
            if (d0 < best[0][i]) { best[0][i] = d0; bidx[0][i] = nA; }
            float d1 = cnA - 2.0f * acc1[i];
            if (d1 < best[1][i]) { best[1][i] = d1; bidx[1][i] = nA; }
            float d2 = cnA - 2.0f * acc2[i];
            if (d2 < best[2][i]) { best[2][i] = d2; bidx[2][i] = nA; }
            float d3 = cnA - 2.0f * acc3[i];
            if (d3 < best[3][i]) { best[3][i] = d3; bidx[3][i] = nA; }
            float d4 = cnB - 2.0f * acc4[i];
            if (d4 < best[0][i]) { best[0][i] = d4; bidx[0][i] = nB; }
            float d5 = cnB - 2.0f * acc5[i];
            if (d5 < best[1][i]) { best[1][i] = d5; bidx[1][i] = nB; }
            float d6 = cnB - 2.0f * acc6[i];
            if (d6 < best[2][i]) { best[2][i] = d6; bidx[2][i] = nB; }
            float d7 = cnB - 2.0f * acc7[i];
            if (d7 < best[3][i]) { best[3][i] = d7; bidx[3][i] = nB; }
        }
    }

    // Cross-lane argmin within each 16-lane half (row set is per-half).
#pragma unroll
    for (int mt = 0; mt < 4; ++mt) {
#pragma unroll
        for (int i = 0; i < 8; ++i) {
            float v = best[mt][i];
            int   ix = bidx[mt][i];
            for (int m = 1; m <= 8; m <<= 1) {
                float ov = __shfl_xor(v, m, 32);
                int   oi = __shfl_xor(ix, m, 32);
                if (ov < v || (ov == v && oi < ix)) { v = ov; ix = oi; }
            }
            if (laneN == 0) {
                int row = mt * 16 + half * 8 + i;   // C layout: VGPR i -> M=i / i+8
                wval[wave * MB + row] = v;
                widx[wave * MB + row] = ix;
            }
        }
    }
    __syncthreads();

    // Reduce across 8 waves, emit per-slice partials.
    if (t < MB) {
        float v = wval[t];
        int   ix = widx[t];
#pragma unroll
        for (int w = 1; w < 8; ++w) {
            float ov = wval[w * MB + t];
            int   oi = widx[w * MB + t];
            if (ov < v || (ov == v && oi < ix)) { v = ov; ix = oi; }
        }
        pval[nsl * NPIX + mbase + t] = v;
        pidx[nsl * NPIX + mbase + t] = ix;
    }
}

// ---------------------------------------------------------------------------
// Kernel 3: reduce the NSLICE partial argmins per pixel.
// ---------------------------------------------------------------------------
__global__ void __launch_bounds__(256) slice_reduce_kernel(
    const float* __restrict__ pval, const int* __restrict__ pidx,
    int* __restrict__ idxf)
{
    int p = blockIdx.x * blockDim.x + threadIdx.x;
    float v = pval[p];
    int   ix = pidx[p];
#pragma unroll
    for (int s = 1; s < NSLICE; ++s) {
        float ov = pval[s * NPIX + p];
        int   oi = pidx[s * NPIX + p];
        if (ov < v || (ov == v && oi < ix)) { v = ov; ix = oi; }
    }
    idxf[p] = ix;
}

// ---------------------------------------------------------------------------
// Kernel 4: 3x3 conv 256->3 (SAME) over quant = codebook[idx], + bias,
// then clamp((y+1)*0.5, 0, 1). Output NCHW [4,3,32,32].
// ---------------------------------------------------------------------------
__global__ void __launch_bounds__(256) conv_kernel(
    const float* __restrict__ cb, const int* __restrict__ idx,
    const float* __restrict__ w, const float* __restrict__ bias,
    float* __restrict__ out)
{
    int t = blockIdx.x * blockDim.x + threadIdx.x;     // 12288 outputs
    if (t >= 4 * 3 * 32 * 32) return;
    int x  = t & 31;
    int y  = (t >> 5) & 31;
    int oc = (t >> 10) % 3;
    int b  = t / (3 * 1024);

    float acc = bias[oc];
    for (int kh = -1; kh <= 1; ++kh) {
        int hh = y + kh;
        if (hh < 0 || hh > 31) continue;
        for (int kw = -1; kw <= 1; ++kw) {
            int ww = x + kw;
            if (ww < 0 || ww > 31) continue;
            int p = b * 1024 + hh * 32 + ww;
            const float4* q = (const float4*)(cb + (size_t)idx[p] * DIM);
            // dec_w[oc][c][kh+1][kw+1] = w[oc*256*9 + c*9 + (kh+1)*3 + (kw+1)]
            const float* wk = w + oc * DIM * 9 + (kh + 1) * 3 + (kw + 1);
            float s = 0.0f;
#pragma unroll 4
            for (int c4 = 0; c4 < DIM / 4; ++c4) {
                float4 v = q[c4];
                int c0 = c4 * 4;
                s += v.x * wk[(c0 + 0) * 9] + v.y * wk[(c0 + 1) * 9]
                   + v.z * wk[(c0 + 2) * 9] + v.w * wk[(c0 + 3) * 9];
            }
            acc += s;
        }
    }
    out[t] = fminf(fmaxf((acc + 1.0f) * 0.5f, 0.0f), 1.0f);
}

// ---------------------------------------------------------------------------
extern "C" void kernel_launch(void* const* d_in, const int* in_sizes, int n_in,
                              void* d_out, int out_size, void* d_ws, size_t ws_size,
                              hipStream_t stream) {
    const float* latent = (const float*)d_in[0];   // [4,256,32,32]
    const float* cb     = (const float*)d_in[1];   // [16384,256]
    const float* decw   = (const float*)d_in[2];   // [3,256,3,3]
    const float* decb   = (const float*)d_in[3];   // [3]
    float* out = (float*)d_out;                    // [4,3,32,32]

    char* ws = (char*)d_ws;
    float* cnorm = (float*)(ws);                   //  64 KB
    float* pval  = (float*)(ws + (64 << 10));      //  64 KB (NSLICE*NPIX f32)
    int*   pidx  = (int*)  (ws + (128 << 10));     //  64 KB
    int*   idxf  = (int*)  (ws + (192 << 10));     //  16 KB

    cnorm_kernel<<<NCODES / 256, 256, 0, stream>>>(cb, cnorm);
    vq_argmin_kernel<<<(NPIX / MB) * NSLICE, 256, 0, stream>>>(latent, cb, cnorm, pval, pidx);
    slice_reduce_kernel<<<NPIX / 256, 256, 0, stream>>>(pval, pidx, idxf);
    conv_kernel<<<(4 * 3 * 32 * 32 + 255) / 256, 256, 0, stream>>>(cb, idxf, decw, decb, out);
}